// sPLL_11785390260243
// MI455X (gfx1250) — compile-verified
//
#include <hip/hip_runtime.h>

// sPLL spiking-network scan for gfx1250 (MI455X).
// One thread = one (trial b, neuron n) cell; T=500 step loop in registers.
// Input column prefetched to LDS via global_load_async_to_lds_b32 (ASYNCcnt),
// spike outputs streamed with non-temporal stores (write-once, 256 MB).

#define TT 500
#define BS 256
#define NN 256

// Get the 32-bit LDS byte offset of a __shared__ object (generic -> local cast).
__device__ __forceinline__ unsigned lds_byte_off(const void* p) {
  return (unsigned)(unsigned long long)(const __attribute__((address_space(3))) char*)p;
}

__global__ __launch_bounds__(256) void spll_scan_kernel(
    const float* __restrict__ inp,   // (T, BS)
    const float* __restrict__ cur,   // (N,)
    float* __restrict__ out) {       // (2, T, BS, N)
  __shared__ __align__(16) float s_in[512];   // padded input column for trial b

  const int n = threadIdx.x;   // neuron index (lane-contiguous -> coalesced stores)
  const int b = blockIdx.x;    // trial index

  // ---- async gather of input[:, b] (stride BS floats) into contiguous LDS ----
  {
    const unsigned t0  = (unsigned)n;          // handles t = n and t = n + 256
    const unsigned t1  = t0 + 256u;
    const unsigned t1c = (t1 < TT) ? t1 : (TT - 1u);  // clamp tail (dup, never read)
    const unsigned base = lds_byte_off(&s_in[0]);
    const unsigned lds0 = base + t0 * 4u;
    const unsigned lds1 = base + t1 * 4u;
    const unsigned off0 = (t0 * BS + (unsigned)b) * 4u;
    const unsigned off1 = (t1c * BS + (unsigned)b) * 4u;
    asm volatile(
        "global_load_async_to_lds_b32 %0, %2, %4\n\t"
        "global_load_async_to_lds_b32 %1, %3, %4"
        :
        : "v"(lds0), "v"(lds1), "v"(off0), "v"(off1), "s"(inp)
        : "memory");
    asm volatile("s_wait_asynccnt 0" ::: "memory");
  }
  __syncthreads();

  // ---- per-cell state in registers ----
  const float steady = cur[n];
  float Vl = 1.0f, Il = 0.0f, crl = 0.0f;                 // LIF oscillator
  float Vt = 0.0f, itrg = 0.0f, ifac = 0.0f, crt = 0.0f;  // TDE
  float spk_prev = 0.0f;

  float* o0 = out + ((size_t)b * NN + n);                 // spk_lif[t,b,n]
  float* o1 = o0 + (size_t)TT * BS * NN;                  // spk_tde[t,b,n]
  const size_t row_stride = (size_t)BS * NN;

  // Constants folded at double precision (match reference values exactly-ish):
  // 1 - DT/TAU_SYN = 0.8, 1 - DT/TAU_FAC = 0.8, 1 - DT/TAU_TRG = 0.9
  // 1/TAU_NEU = 1/BETA = 100, 1/C_OSC = 1/C_TDE = 1000, REFR = 2.0 steps
  for (int t = 0; t < TT; t += 4) {
    // uniform broadcast read: one ds_load_b128 per 4 steps
    const float4 xv = *(const float4*)(&s_in[t]);
    const float xs[4] = {xv.x, xv.y, xv.z, xv.w};
#pragma unroll
    for (int j = 0; j < 4; ++j) {
      const float x = xs[j];

      // ---- LIF step (driven by previous TDE spikes) ----
      Il = Il * 0.8f + 1e-3f * spk_prev;
      Vl = Vl + 1e-3f * (Vl * -100.0f + (Il + steady) * 1000.0f);
      const float spk_l = (Vl > 1.0f) ? 1.0f : 0.0f;
      crl = (spk_l != 0.0f) ? 2.0f : (crl - 1.0f);
      Vl = ((spk_l == 0.0f) && (crl <= 0.0f)) ? Vl : 0.0f;   // VR_OSC = 0

      // ---- TDE step (trg = external input, fac = LIF spikes) ----
      ifac = (spk_l != 0.0f) ? 1.0f : ifac;   // GAIN_FAC = 1
      ifac = ifac * 0.8f;
      itrg = itrg * 0.9f;
      itrg = itrg + ((ifac > 0.1f) ? (x * 2.0f * ifac) : 0.0f);  // GAIN_TRG = 2
      Vt = Vt + 1e-3f * (Vt * -100.0f + itrg * 1000.0f);
      const float spk_t = (Vt > 1.0f) ? 1.0f : 0.0f;
      crt = (spk_t != 0.0f) ? 2.0f : (crt - 1.0f);
      Vt = ((spk_t == 0.0f) && (crt <= 0.0f)) ? Vt : 0.0f;   // VR_TDE = 0
      spk_prev = spk_t;

      // ---- streaming (non-temporal) spike outputs ----
      const size_t row = (size_t)(t + j) * row_stride;
      __builtin_nontemporal_store(spk_l, o0 + row);
      __builtin_nontemporal_store(spk_t, o1 + row);
    }
  }
}

extern "C" void kernel_launch(void* const* d_in, const int* in_sizes, int n_in,
                              void* d_out, int out_size, void* d_ws, size_t ws_size,
                              hipStream_t stream) {
  (void)in_sizes; (void)n_in; (void)out_size; (void)d_ws; (void)ws_size;
  const float* inp = (const float*)d_in[0];   // (T, BS) float32
  const float* cur = (const float*)d_in[1];   // (N,)  float32
  float* out = (float*)d_out;                 // (2, T, BS, N) float32

  spll_scan_kernel<<<dim3(BS), dim3(NN), 0, stream>>>(inp, cur, out);
}